// Sn_coarsen_layer_reflect_6296422056038
// MI455X (gfx1250) — compile-verified
//
#include <hip/hip_runtime.h>
#include <hip/hip_bf16.h>

typedef __attribute__((ext_vector_type(2))) float v2f;
typedef __attribute__((ext_vector_type(8))) float v8f;

#define BSZ 32
#define NN  256
#define INC 32
#define OUTC 32

// ---------------------------------------------------------------------------
// Kernel 1: group sums of permuted x, pre-scaled by 1/256, stored K-pair-major:
//   Yp[(k>>1)*64 + b*2 + (k&1)] = (1/256) * sum_{r<8} x[b, perm[8g+r], i]
// with k = g*32 + i.  Makes the WMMA A-fragment a single coalesced b64.
// grid = 32 (b), block = 256
// ---------------------------------------------------------------------------
__global__ __launch_bounds__(256) void group_sum_kernel(
    const float* __restrict__ x,      // [32][256][32]
    const int*   __restrict__ perm,   // [256]
    float*       __restrict__ Yp)     // [512][32][2] f32 (k-pair, b, k&1)
{
    __shared__ int p[256];
    const int b   = blockIdx.x;
    const int tid = threadIdx.x;
    p[tid] = perm[tid];
    __syncthreads();

    const float SCALE = 1.0f / 256.0f;
    #pragma unroll
    for (int r = 0; r < 4; ++r) {
        const int e = tid + r * 256;       // k = 0..1023
        const int g = e >> 5;
        const int i = e & 31;
        float s = 0.f;
        #pragma unroll
        for (int q = 0; q < 8; ++q)
            s += x[(b * NN + p[g * 8 + q]) * INC + i];
        Yp[(e >> 1) * 64 + b * 2 + (e & 1)] = s * SCALE;
    }
}

// ---------------------------------------------------------------------------
// Kernel 2: per-h GEMM  Z[b,h,o] = sum_k Yp[b,k] * Bh[k,o]   (scale pre-folded)
//   Bh[k,o] = w_off[(o*32 + (k&31))*512 + (h>>1)*32 + ((k>>5) ^ (h&1))]
// grid = 32 (h), block = 256 = 8 waves; wave w owns K range [w*128, w*128+128).
// Bh (128 KB) staged into LDS K-pair-packed so every B fragment is ONE
// aligned ds_load_b64 (no register shuffles into the even-aligned WMMA pair).
// The 32 KB wave-partial reduction buffer aliases the same LDS afterwards.
// ---------------------------------------------------------------------------
__global__ __launch_bounds__(256) void gemm_h_kernel(
    const float* __restrict__ Yp,     // [512][32][2]
    const float* __restrict__ w_off,  // [1024][16][32]
    float*       __restrict__ Z)      // [32][32][32]  (b,h,o)
{
    __shared__ float smem[1024 * 32]; // 128 KB: Bp[k>>1][o][k&1] during K loop,
                                      // first 8K floats reused for reduction
    const int h    = blockIdx.x;      // 0..31
    const int tid  = threadIdx.x;
    const int wave = tid >> 5;        // 0..7
    const int lane = tid & 31;
    const int l15  = lane & 15;
    const int lhl  = lane >> 4;       // 0/1 half of wave

    const int hk = h >> 1;            // w_off middle index
    const int hx = h & 1;             // xor for pair swap

    // ---- stage Bh into LDS, fully coalesced float4 reads of w_off ----
    // LDS layout: smem[(k>>1)*64 + o*2 + (k&1)], k = g*32+i  (k&1 == i&1)
    __builtin_prefetch(w_off + (size_t)tid * 512, 0, 3);
    #pragma unroll 4
    for (int r = 0; r < 32; ++r) {
        const int f  = tid + r * 256;          // 0..8191 float4 chunks
        const int c  = f >> 3;                 // 0..1023  (c = o*32 + i)
        const int g4 = f & 7;                  // float4 chunk within row
        const float4 v = ((const float4*)(w_off + c * 512 + hk * 32))[g4];
        const int o  = c >> 5;
        const int i  = c & 31;
        const int base = (i >> 1) * 64 + o * 2 + (i & 1); // k-pair packing of i
        const int gb = (g4 << 2);
        // k>>1 = g'*16 + (i>>1)  with g' = g ^ hx
        smem[(((gb + 0) ^ hx) << 10) + base] = v.x;   // g'*16*64 = g'<<10
        smem[(((gb + 1) ^ hx) << 10) + base] = v.y;
        smem[(((gb + 2) ^ hx) << 10) + base] = v.z;
        smem[(((gb + 3) ^ hx) << 10) + base] = v.w;
    }
    __syncthreads();

    // ---- K loop: 2x2 tiles of V_WMMA_F32_16X16X4_F32 ----
    v8f c[2][2] = {};
    const int kbase = wave * 128;
    const int o0 = l15;
    const int o1 = 16 + l15;

    for (int s = 0; s < 32; ++s) {
        const int kl = kbase + s * 4 + lhl * 2;  // this lane's first K (even)
        const int pp = kl >> 1;                  // K-pair index

        // A fragments: one coalesced b64 per M-tile
        v2f a[2];
        #pragma unroll
        for (int tm = 0; tm < 2; ++tm) {
            const float2 af =
                ((const float2*)Yp)[pp * 32 + tm * 16 + l15];
            a[tm].x = af.x;
            a[tm].y = af.y;
        }

        // B fragments: one aligned ds_load_b64 each
        const float2 bl0 = ((const float2*)smem)[pp * 32 + o0];
        const float2 bl1 = ((const float2*)smem)[pp * 32 + o1];
        v2f bf0, bf1;
        bf0.x = bl0.x; bf0.y = bl0.y;
        bf1.x = bl1.x; bf1.y = bl1.y;

        c[0][0] = __builtin_amdgcn_wmma_f32_16x16x4_f32(
            false, a[0], false, bf0, (short)0, c[0][0], false, false);
        c[0][1] = __builtin_amdgcn_wmma_f32_16x16x4_f32(
            false, a[0], false, bf1, (short)0, c[0][1], false, false);
        c[1][0] = __builtin_amdgcn_wmma_f32_16x16x4_f32(
            false, a[1], false, bf0, (short)0, c[1][0], false, false);
        c[1][1] = __builtin_amdgcn_wmma_f32_16x16x4_f32(
            false, a[1], false, bf1, (short)0, c[1][1], false, false);
    }

    // ---- reduce the 8 wave partials through (aliased) LDS ----
    __syncthreads();   // everyone done reading Bp
    #pragma unroll
    for (int tm = 0; tm < 2; ++tm)
        #pragma unroll
        for (int tn = 0; tn < 2; ++tn)
            #pragma unroll
            for (int v = 0; v < 8; ++v)
                smem[wave * 1024 + (tm * 2 + tn) * 256 + v * 32 + lane] =
                    c[tm][tn][v];
    __syncthreads();

    #pragma unroll
    for (int r = 0; r < 4; ++r) {
        const int e = tid + r * 256;           // 0..1023
        float sum = 0.f;
        #pragma unroll
        for (int w = 0; w < 8; ++w)
            sum += smem[w * 1024 + e];
        // decode (tile, vgpr, lane) -> (m=b, n=o) per ISA C/D f32 layout
        const int tt = e >> 8;                 // tile = tm*2+tn
        const int v  = (e >> 5) & 7;
        const int l  = e & 31;
        const int m  = (tt >> 1) * 16 + v + 8 * (l >> 4);   // b
        const int n  = (tt & 1) * 16 + (l & 15);            // o
        Z[(m * 32 + h) * 32 + n] = sum;
    }
}

// ---------------------------------------------------------------------------
// Kernel 3: out[b,n,o] = b1[o] + Z[b, s>>3, o] + sum_i w_diag[o*32+i, s>>4]*x[b,n,i]
//           with s = inv_perm[n].
// grid = 32*32 (b, n-chunk of 8), block = 256: thread = (n_local*32 + o)
// ---------------------------------------------------------------------------
__global__ __launch_bounds__(256) void final_kernel(
    const float* __restrict__ x,        // [32][256][32]
    const float* __restrict__ w_diag,   // [1024][16]
    const float* __restrict__ b1,       // [32]
    const int*   __restrict__ inv_perm, // [256]
    const float* __restrict__ Z,        // [32][32][32]
    float*       __restrict__ out)      // [32][256][32]
{
    const int b   = blockIdx.x >> 5;
    const int nc  = blockIdx.x & 31;
    const int tid = threadIdx.x;
    const int nl  = tid >> 5;            // 0..7
    const int o   = tid & 31;
    const int n   = nc * 8 + nl;

    __shared__ float xr[8][32];
    __shared__ int   sp[8];
    xr[nl][o] = x[(b * NN + n) * INC + o];
    if (o == 0) sp[nl] = inv_perm[n];
    __syncthreads();

    const int s  = sp[nl];
    const int sd = s >> 4;
    float acc = b1[o] + Z[(b * 32 + (s >> 3)) * 32 + o];

    const float* wd = w_diag + o * 512 + sd;   // w_diag[(o*32+i)*16 + sd]
    #pragma unroll
    for (int i = 0; i < 32; ++i)
        acc += wd[i * 16] * xr[nl][i];

    out[(b * NN + n) * INC + o] = acc;
}

// ---------------------------------------------------------------------------
extern "C" void kernel_launch(void* const* d_in, const int* in_sizes, int n_in,
                              void* d_out, int out_size, void* d_ws, size_t ws_size,
                              hipStream_t stream) {
    (void)in_sizes; (void)n_in; (void)out_size; (void)ws_size;

    const float* x        = (const float*)d_in[0];  // 32*256*32
    const float* w_diag   = (const float*)d_in[1];  // 1024*16
    const float* w_off    = (const float*)d_in[2];  // 1024*16*32
    const float* b1       = (const float*)d_in[3];  // 32
    const int*   perm     = (const int*)d_in[4];    // 256
    const int*   inv_perm = (const int*)d_in[5];    // 256
    float*       out      = (float*)d_out;          // 32*256*32

    float* Yp = (float*)d_ws;                       // 32*1024 f32 = 128 KB
    float* Z  = Yp + 32 * 1024;                     // 32*32*32 f32 = 128 KB

    group_sum_kernel<<<32, 256, 0, stream>>>(x, perm, Yp);
    gemm_h_kernel<<<32, 256, 0, stream>>>(Yp, w_off, Z);
    final_kernel<<<32 * 32, 256, 0, stream>>>(x, w_diag, b1, inv_perm, Z, out);
}